// EnDLoss_37726992728790
// MI455X (gfx1250) — compile-verified
//
#include <hip/hip_runtime.h>
#include <hip/hip_bf16.h>
#include <stdint.h>

// ---------------------------------------------------------------------------
// EnD-style loss on MI455X (gfx1250):
//   label_loss = mean CE(logits, labels)            [tiny]
//   bias_loss  = R_ortho + R_parallel over gram = clamp(tril(F F^T), -1, 1)
// Gram (4096x4096 from 4096x2048 f32) is never materialized: lower-tri
// 128x128 tiles are computed with V_WMMA_F32_16X16X4_F32 from TDM-staged,
// double-buffered LDS tiles (tensor_load_to_lds overlapped with WMMA via
// in-order TENSORcnt) and reduced in-register to 4 scalars.
// ---------------------------------------------------------------------------

typedef __attribute__((ext_vector_type(2)))  float        v2f;
typedef __attribute__((ext_vector_type(8)))  float        v8f;
typedef __attribute__((ext_vector_type(4)))  unsigned int v4u;
typedef __attribute__((ext_vector_type(8)))  int          v8i;
typedef __attribute__((ext_vector_type(4)))  int          v4i;

#define BM 128
#define BN 128
#define BK 32
#define LDS_STRIDE 33   // 32 data dwords + 1 TDM pad dword per row -> no bank conflicts
#define TILE_BYTES (BM * LDS_STRIDE * 4)   // 16896

struct SMem {
  float A[2][BM][LDS_STRIDE];   // LDS offsets 0, 16896       (i-block rows, double buffered)
  float B[2][BN][LDS_STRIDE];   // LDS offsets 33792, 50688   (j-block rows, double buffered)
  int   labI[BM];
  int   biaI[BM];
  int   labJ[BN];
  int   biaJ[BN];
  float red[4];                 // S0, S1, psum, pcount block partials
};

// Workspace layout (floats): [0]=S0 [1]=S1 [2]=psum [3]=pcount [4]=ce_sum [5]=n0 [6]=n1
#define WS_S0 0
#define WS_S1 1
#define WS_PS 2
#define WS_PC 3
#define WS_CE 4
#define WS_N0 5
#define WS_N1 6

// --- Tensor Data Mover: 2D tile load (global -> LDS), f32 elements, with
// --- per-32-dword LDS padding of 1 dword (pad_interval code 4, pad_amount 0).
__device__ __forceinline__ void tdm_load_2d(unsigned lds_off,
                                            unsigned long long gaddr,
                                            unsigned tensor_d0,
                                            unsigned tensor_d1,
                                            unsigned long long d0_stride,
                                            unsigned tile_d0,
                                            unsigned tile_d1) {
  v4u g0;
  g0.x = 1u;                                           // count=1, user descriptor
  g0.y = lds_off;                                      // lds_addr (bytes)
  g0.z = (unsigned)(gaddr & 0xFFFFFFFFull);            // global_addr[31:0]
  g0.w = (unsigned)((gaddr >> 32) & 0x01FFFFFFull)     // global_addr[56:32]
       | (2u << 30);                                   // type = 2 ("image")

  v8i g1;
  unsigned w0 = (2u << 16)    // data_size = 4 bytes
              | (1u << 20)    // pad_enable
              | (4u << 22)    // pad_interval: every 32 dwords
              | (0u << 25);   // pad_amount: 1 dword
  g1[0] = (int)w0;                                                        // mask=0 (not in cluster)
  g1[1] = (int)((tensor_d0 & 0xFFFFu) << 16);                             // dim0[15:0] @ bits63:48
  g1[2] = (int)(((tensor_d0 >> 16) & 0xFFFFu) | ((tensor_d1 & 0xFFFFu) << 16));
  g1[3] = (int)(((tensor_d1 >> 16) & 0xFFFFu) | ((tile_d0  & 0xFFFFu) << 16));
  g1[4] = (int)(tile_d1 & 0xFFFFu);                                       // tile_dim2 = 0 (2D)
  g1[5] = (int)(unsigned)(d0_stride & 0xFFFFFFFFull);                     // dim0_stride[31:0]
  g1[6] = (int)(unsigned)((d0_stride >> 32) & 0xFFFFull);                 // dim0_stride[47:32]
  g1[7] = 0;                                                              // dim1_stride unused (2D)

  v4i g2 = {0, 0, 0, 0};
  v4i g3 = {0, 0, 0, 0};
  v8i g4 = {0, 0, 0, 0, 0, 0, 0, 0};   // extra group (clang-23 6-arg form), unused for 2D
  __builtin_amdgcn_tensor_load_to_lds(g0, g1, g2, g3, g4, 0);
}

// --- Kernel 0: zero the accumulator workspace (deterministic each launch) ---
__global__ void ws_init_kernel(float* __restrict__ ws) {
  if (threadIdx.x < 7) ws[threadIdx.x] = 0.0f;
}

// --- Kernel 1: cross entropy partial sum + per-bias-class counts ---
__global__ __launch_bounds__(256)
void ce_kernel(const float* __restrict__ logits,
               const int* __restrict__ labels,
               const int* __restrict__ biases,
               float* __restrict__ ws, int N, int C) {
  int i = blockIdx.x * blockDim.x + threadIdx.x;
  if (i >= N) return;
  const float* row = logits + (size_t)i * C;
  float m = row[0];
  for (int c = 1; c < C; ++c) m = fmaxf(m, row[c]);
  float s = 0.0f;
  for (int c = 0; c < C; ++c) s += expf(row[c] - m);
  float loss = logf(s) + m - row[labels[i]];
  atomicAdd(&ws[WS_CE], loss);
  atomicAdd(&ws[WS_N0 + (biases[i] & 1)], 1.0f);
}

// --- Kernel 2: fused lower-triangular gram tiles + masked reductions ---
__global__ __launch_bounds__(256)
void gram_kernel(const float* __restrict__ feats,
                 const int* __restrict__ labels,
                 const int* __restrict__ biases,
                 float* __restrict__ ws, int N, int D) {
  __shared__ SMem sm;
  const int tid = threadIdx.x;

  // Map linear block id -> lower-triangular (bi >= bj) 128x128 block coords.
  int t  = blockIdx.x;
  int bi = (int)((sqrtf(8.0f * (float)t + 1.0f) - 1.0f) * 0.5f);
  while ((bi + 1) * (bi + 2) / 2 <= t) ++bi;
  while (bi * (bi + 1) / 2 > t) --bi;
  const int bj = t - bi * (bi + 1) / 2;
  const int i0 = bi * BM;
  const int j0 = bj * BN;

  // Stage block metadata + zero reduction scratch.
  if (tid < BM) { sm.labI[tid] = labels[i0 + tid]; sm.biaI[tid] = biases[i0 + tid]; }
  else if (tid < 2 * BM) {
    int j = tid - BM;
    sm.labJ[j] = labels[j0 + j]; sm.biaJ[j] = biases[j0 + j];
  }
  if (tid < 4) sm.red[tid] = 0.0f;

  const int wave = tid >> 5;
  const int lane = tid & 31;
  const int hlf  = lane >> 4;   // 0: lanes 0-15, 1: lanes 16-31
  const int l16  = lane & 15;
  const int wm   = (wave & 3) * 32;   // 4 waves down  (2 M-subtiles each)
  const int wn   = (wave >> 2) * 64;  // 2 waves across (4 N-subtiles each)

  v8f acc[2][4];
  #pragma unroll
  for (int tm = 0; tm < 2; ++tm)
    #pragma unroll
    for (int tn = 0; tn < 4; ++tn)
      acc[tm][tn] = v8f{0.f, 0.f, 0.f, 0.f, 0.f, 0.f, 0.f, 0.f};

  const unsigned long long base  = (unsigned long long)(uintptr_t)feats;
  const unsigned long long rowA0 = base + (unsigned long long)i0 * D * 4ull;
  const unsigned long long rowB0 = base + (unsigned long long)j0 * D * 4ull;
  const int nstage = D / BK;

  // Prologue: wave 0 kicks off stage 0 DMA into buffer 0.
  if (wave == 0) {
    tdm_load_2d(0u * TILE_BYTES, rowA0, (unsigned)D, (unsigned)N,
                (unsigned long long)D, BK, BM);
    tdm_load_2d(2u * TILE_BYTES, rowB0, (unsigned)D, (unsigned)N,
                (unsigned long long)D, BK, BN);
  }

  for (int s = 0; s < nstage; ++s) {
    const int buf = s & 1;
    // All waves are done computing on buf^1 (stage s-1) -> safe to refill it.
    __syncthreads();
    if (wave == 0) {
      if (s + 1 < nstage) {
        const unsigned long long koff = (unsigned long long)(s + 1) * BK * 4ull;
        const unsigned nb = (unsigned)(buf ^ 1);
        tdm_load_2d(nb * TILE_BYTES, rowA0 + koff, (unsigned)D, (unsigned)N,
                    (unsigned long long)D, BK, BM);
        tdm_load_2d((2u + nb) * TILE_BYTES, rowB0 + koff, (unsigned)D, (unsigned)N,
                    (unsigned long long)D, BK, BN);
        // TENSORcnt retires in order: <=2 outstanding means stage s is complete,
        // only the two stage-(s+1) transfers remain in flight.
        __builtin_amdgcn_s_wait_tensorcnt(2);
      } else {
        __builtin_amdgcn_s_wait_tensorcnt(0);
      }
    }
    __syncthreads();  // stage-s data visible to all 8 waves; DMA of s+1 overlaps below

    #pragma unroll
    for (int k = 0; k < BK; k += 4) {
      // A frag (16x4 f32): v0 = K(2*hlf), v1 = K(2*hlf+1), rows on lanes%16
      v2f af[2], bf[4];
      #pragma unroll
      for (int tm = 0; tm < 2; ++tm) {
        const int row = wm + tm * 16 + l16;
        af[tm].x = sm.A[buf][row][k + 2 * hlf];
        af[tm].y = sm.A[buf][row][k + 2 * hlf + 1];
      }
      // B frag (4x16 f32): B[k'][n] = feats[j0+n][k'] -> read sm.B transposed
      #pragma unroll
      for (int tn = 0; tn < 4; ++tn) {
        const int col = wn + tn * 16 + l16;
        bf[tn].x = sm.B[buf][col][k + 2 * hlf];
        bf[tn].y = sm.B[buf][col][k + 2 * hlf + 1];
      }
      #pragma unroll
      for (int tm = 0; tm < 2; ++tm)
        #pragma unroll
        for (int tn = 0; tn < 4; ++tn)
          acc[tm][tn] = __builtin_amdgcn_wmma_f32_16x16x4_f32(
              false, af[tm], false, bf[tn], (short)0, acc[tm][tn], false, false);
    }
  }

  // Epilogue: clamp + masks + per-lane partial reductions.
  // C/D layout: VGPR v -> M = v + 8*hlf, N = lane%16.
  float s0 = 0.f, s1 = 0.f, ps = 0.f, pc = 0.f;
  #pragma unroll
  for (int tm = 0; tm < 2; ++tm) {
    #pragma unroll
    for (int tn = 0; tn < 4; ++tn) {
      const int jl  = wn + tn * 16 + l16;
      const int lj  = sm.labJ[jl];
      const int bjv = sm.biaJ[jl];
      #pragma unroll
      for (int v = 0; v < 8; ++v) {
        const int il = wm + tm * 16 + v + 8 * hlf;
        const int gi = i0 + il;
        const int gj = j0 + jl;
        if (gi > gj) {  // strict lower triangle
          const float g = fminf(fmaxf(acc[tm][tn][v], -1.0f), 1.0f);
          const int li  = sm.labI[il];
          const int biv = sm.biaI[il];
          if (biv == bjv) { if (biv == 0) s0 += g; else s1 += g; }
          if (li == lj && biv < bjv) { ps += 1.0f + g; pc += 1.0f; }
        }
      }
    }
  }

  atomicAdd(&sm.red[0], s0);
  atomicAdd(&sm.red[1], s1);
  atomicAdd(&sm.red[2], ps);
  atomicAdd(&sm.red[3], pc);
  __syncthreads();
  if (tid == 0) {
    atomicAdd(&ws[WS_S0], sm.red[0]);
    atomicAdd(&ws[WS_S1], sm.red[1]);
    atomicAdd(&ws[WS_PS], sm.red[2]);
    atomicAdd(&ws[WS_PC], sm.red[3]);
  }
}

// --- Kernel 3: combine partials into (label_loss, bias_loss) ---
__global__ void finalize_kernel(const float* __restrict__ ws,
                                float* __restrict__ out, float fN) {
  const float label_loss = ws[WS_CE] / fN;
  const float n0 = ws[WS_N0], n1 = ws[WS_N1];
  const float Mt = n0 * (n0 - 1.0f) * 0.5f + n1 * (n1 - 1.0f) * 0.5f;
  const float R_ortho = (Mt > 0.0f) ? (fabsf(ws[WS_S0]) + fabsf(ws[WS_S1])) / Mt : 0.0f;
  const float pc = ws[WS_PC];
  const float R_par = (pc > 0.0f) ? (1.0f + (-0.5f * ws[WS_PS]) / pc) : 0.0f;
  out[0] = label_loss;
  out[1] = R_ortho + R_par;  // ALPHA = BETA = 1
}

extern "C" void kernel_launch(void* const* d_in, const int* in_sizes, int n_in,
                              void* d_out, int out_size, void* d_ws, size_t ws_size,
                              hipStream_t stream) {
  const float* logits = (const float*)d_in[0];
  const int*   labels = (const int*)d_in[1];
  const int*   biases = (const int*)d_in[2];
  const float* feats  = (const float*)d_in[3];
  float* out = (float*)d_out;
  float* ws  = (float*)d_ws;

  const int N = in_sizes[1];
  const int C = in_sizes[0] / N;
  const int D = in_sizes[3] / N;

  ws_init_kernel<<<1, 32, 0, stream>>>(ws);
  ce_kernel<<<(N + 255) / 256, 256, 0, stream>>>(logits, labels, biases, ws, N, C);

  const int nb = N / BM;                 // 4096/128 = 32
  const int ntiles = nb * (nb + 1) / 2;  // 528 lower-tri blocks
  gram_kernel<<<ntiles, 256, 0, stream>>>(feats, labels, biases, ws, N, D);

  finalize_kernel<<<1, 1, 0, stream>>>(ws, out, (float)N);
}